// DORALayer_4243427688518
// MI455X (gfx1250) — compile-verified
//
#include <hip/hip_runtime.h>
#include <math.h>

typedef __attribute__((ext_vector_type(2))) float v2f;
typedef __attribute__((ext_vector_type(8))) float v8f;

#define IN_DIM   4096
#define OUT_DIM  4096
#define RANK     64
#define NROWS    8192      /* B * S = 4 * 2048 */
#define NORM_SLICES 8

// D = A(16x4 f32) * B(4x16 f32) + C(16x16 f32), full fp32 precision.
__device__ __forceinline__ v8f wmma_f32(v2f a, v2f b, v8f c) {
  return __builtin_amdgcn_wmma_f32_16x16x4_f32(
      /*neg_a=*/false, a, /*neg_b=*/false, b,
      /*c_mod=*/(short)0, c, /*reuse_a=*/false, /*reuse_b=*/false);
}

// ---------------------------------------------------------------------------
// Kernel 1: h[m, r] = sum_i x[m, i] * Aw[r, i]     (M=8192, K=4096, N=64)
// One wave per 16-row m-tile; all 4 rank-tiles accumulated in registers so
// x (the big operand, 134 MB) is streamed from HBM exactly once.
// ---------------------------------------------------------------------------
__global__ void __launch_bounds__(256) lora_h_kernel(
    const float* __restrict__ x, const float* __restrict__ Aw,
    float* __restrict__ h)
{
  const int wave = blockIdx.x * (blockDim.x >> 5) + (threadIdx.x >> 5);
  const int m0 = wave * 16;
  if (m0 >= NROWS) return;
  const int lane = threadIdx.x & 31;
  const int lo = lane & 15;      // M (A-frag) / N (B-frag) index within tile
  const int hi = lane >> 4;      // selects K pair {0,1} vs {2,3}

  const float* xr  = x  + (size_t)(m0 + lo) * IN_DIM + 2 * hi;
  const float* ar0 = Aw + (size_t)(lo      ) * IN_DIM + 2 * hi;
  const float* ar1 = Aw + (size_t)(lo + 16) * IN_DIM + 2 * hi;
  const float* ar2 = Aw + (size_t)(lo + 32) * IN_DIM + 2 * hi;
  const float* ar3 = Aw + (size_t)(lo + 48) * IN_DIM + 2 * hi;

  v8f acc0 = {}, acc1 = {}, acc2 = {}, acc3 = {};
#pragma unroll 4
  for (int k = 0; k < IN_DIM; k += 4) {
    v2f a  = *(const v2f*)(xr  + k);
    v2f b0 = *(const v2f*)(ar0 + k);
    v2f b1 = *(const v2f*)(ar1 + k);
    v2f b2 = *(const v2f*)(ar2 + k);
    v2f b3 = *(const v2f*)(ar3 + k);
    acc0 = wmma_f32(a, b0, acc0);
    acc1 = wmma_f32(a, b1, acc1);
    acc2 = wmma_f32(a, b2, acc2);
    acc3 = wmma_f32(a, b3, acc3);
  }

  float* ho = h + (size_t)m0 * RANK;
#pragma unroll
  for (int j = 0; j < 8; ++j) {
    const size_t row = (size_t)(j + 8 * hi) * RANK;
    ho[row +      lo] = acc0[j];
    ho[row + 16 + lo] = acc1[j];
    ho[row + 32 + lo] = acc2[j];
    ho[row + 48 + lo] = acc3[j];
  }
}

// ---------------------------------------------------------------------------
// Kernel 2: out[m, n] = sum_r h[m, r] * Bw[n, r]   (M=8192, K=64, N=4096)
// Wave per (m-tile, 64-wide n-group): 4 accumulators, 16 K-steps.
// ---------------------------------------------------------------------------
__global__ void __launch_bounds__(256) lora_out_kernel(
    const float* __restrict__ h, const float* __restrict__ Bw,
    float* __restrict__ out)
{
  const int NGRP = OUT_DIM / 64;   // 64 n-groups
  const int wave = blockIdx.x * (blockDim.x >> 5) + (threadIdx.x >> 5);
  const int m0 = (wave / NGRP) * 16;
  const int n0 = (wave % NGRP) * 64;
  if (m0 >= NROWS) return;
  const int lane = threadIdx.x & 31;
  const int lo = lane & 15;
  const int hi = lane >> 4;

  const float* hr  = h  + (size_t)(m0 + lo) * RANK + 2 * hi;
  const float* b0r = Bw + (size_t)(n0      + lo) * RANK + 2 * hi;
  const float* b1r = Bw + (size_t)(n0 + 16 + lo) * RANK + 2 * hi;
  const float* b2r = Bw + (size_t)(n0 + 32 + lo) * RANK + 2 * hi;
  const float* b3r = Bw + (size_t)(n0 + 48 + lo) * RANK + 2 * hi;

  v8f acc0 = {}, acc1 = {}, acc2 = {}, acc3 = {};
#pragma unroll
  for (int s = 0; s < RANK; s += 4) {
    v2f a = *(const v2f*)(hr + s);
    acc0 = wmma_f32(a, *(const v2f*)(b0r + s), acc0);
    acc1 = wmma_f32(a, *(const v2f*)(b1r + s), acc1);
    acc2 = wmma_f32(a, *(const v2f*)(b2r + s), acc2);
    acc3 = wmma_f32(a, *(const v2f*)(b3r + s), acc3);
  }

  float* orow = out + (size_t)m0 * OUT_DIM + n0;
#pragma unroll
  for (int j = 0; j < 8; ++j) {
    const size_t off = (size_t)(j + 8 * hi) * OUT_DIM;
    orow[off +      lo] = acc0[j];   // lanes 0-15 / 16-31 -> two 64B segments
    orow[off + 16 + lo] = acc1[j];
    orow[off + 32 + lo] = acc2[j];
    orow[off + 48 + lo] = acc3[j];
  }
}

// ---------------------------------------------------------------------------
// Kernel 3: partial[slice][o] = sum_{i in slice} (Fw[o,i] + (B@A)[o,i])^2
// Wave per (16-row o-tile, IN-slice). lora_B fragments preloaded (loop
// invariant); 16 WMMAs rebuild each 16x16 B@A tile (K = RANK = 64).
// Deterministic: no float atomics — each (slice, o) written exactly once.
// ---------------------------------------------------------------------------
__global__ void __launch_bounds__(256) dora_norm_partial_kernel(
    const float* __restrict__ Fw, const float* __restrict__ Aw,
    const float* __restrict__ Bw, float* __restrict__ partial)
{
  const int wave  = blockIdx.x * (blockDim.x >> 5) + (threadIdx.x >> 5);
  const int otile = wave / NORM_SLICES;
  const int slice = wave % NORM_SLICES;
  const int o0 = otile * 16;
  if (o0 >= OUT_DIM) return;
  const int lane = threadIdx.x & 31;
  const int lo = lane & 15;
  const int hi = lane >> 4;

  // Preload lora_B A-fragments: rows o0..o0+15, all K = 0..63.
  v2f bf[16];
#pragma unroll
  for (int s = 0; s < 16; ++s)
    bf[s] = *(const v2f*)(Bw + (size_t)(o0 + lo) * RANK + 4 * s + 2 * hi);

  const int i_begin = slice * (IN_DIM / NORM_SLICES);
  const int i_end   = i_begin + (IN_DIM / NORM_SLICES);

  v8f sq = {};
  for (int i0 = i_begin; i0 < i_end; i0 += 16) {
    v8f t = {};
#pragma unroll
    for (int s = 0; s < 16; ++s) {
      const int k = 4 * s + 2 * hi;
      v2f b;
      b.x = Aw[(size_t)k       * IN_DIM + i0 + lo];
      b.y = Aw[(size_t)(k + 1) * IN_DIM + i0 + lo];
      t = wmma_f32(bf[s], b, t);
    }
#pragma unroll
    for (int j = 0; j < 8; ++j) {
      const float m = t[j] + Fw[(size_t)(o0 + j + 8 * hi) * IN_DIM + i0 + lo];
      sq[j] += m * m;
    }
  }

  // Reduce across the 16 lanes of each half-wave (xor masks < 16 keep 'hi').
#pragma unroll
  for (int msk = 8; msk >= 1; msk >>= 1) {
#pragma unroll
    for (int j = 0; j < 8; ++j)
      sq[j] += __shfl_xor(sq[j], msk, 32);
  }
  if (lo == 0) {
#pragma unroll
    for (int j = 0; j < 8; ++j)
      partial[(size_t)slice * OUT_DIM + o0 + j + 8 * hi] = sq[j];
  }
}

// ---------------------------------------------------------------------------
// Kernel 4: column_norm[o] = sqrt(sum_slices partial[slice][o])
// ---------------------------------------------------------------------------
__global__ void __launch_bounds__(256) dora_norm_finalize_kernel(
    const float* __restrict__ partial, float* __restrict__ norm_out)
{
  const int o = blockIdx.x * blockDim.x + threadIdx.x;
  if (o >= OUT_DIM) return;
  float s = 0.f;
#pragma unroll
  for (int p = 0; p < NORM_SLICES; ++p)
    s += partial[(size_t)p * OUT_DIM + o];
  norm_out[o] = sqrtf(s);
}

extern "C" void kernel_launch(void* const* d_in, const int* in_sizes, int n_in,
                              void* d_out, int out_size, void* d_ws, size_t ws_size,
                              hipStream_t stream) {
  (void)in_sizes; (void)n_in; (void)out_size; (void)ws_size;
  const float* x  = (const float*)d_in[0];   // [4, 2048, 4096]
  const float* Fw = (const float*)d_in[1];   // [4096, 4096]
  const float* Aw = (const float*)d_in[2];   // [64, 4096]
  const float* Bw = (const float*)d_in[3];   // [4096, 64]

  float* out      = (float*)d_out;                       // [8192*4096] output
  float* norm_out = out + (size_t)NROWS * OUT_DIM;       // [4096] column_norm

  float* h        = (float*)d_ws;                        // [8192*64]  (2 MB)
  float* partial  = h + (size_t)NROWS * RANK;            // [8][4096]  (128 KB)

  // h = x @ A^T  : 512 waves, 8 waves/block
  lora_h_kernel<<<(NROWS / 16) / 8, 256, 0, stream>>>(x, Aw, h);

  // DoRA norm partials : 2048 waves (independent of h; stream-ordered)
  dora_norm_partial_kernel<<<((OUT_DIM / 16) * NORM_SLICES) / 8, 256, 0, stream>>>(
      Fw, Aw, Bw, partial);

  // out = h @ B^T : 32768 waves
  lora_out_kernel<<<((NROWS / 16) * (OUT_DIM / 64)) / 8, 256, 0, stream>>>(
      h, Bw, out);

  // column_norm = sqrt(sum partials)
  dora_norm_finalize_kernel<<<OUT_DIM / 256, 256, 0, stream>>>(partial, norm_out);
}